// S4Forecast_23940147708105
// MI455X (gfx1250) — compile-verified
//
#include <hip/hip_runtime.h>
#include <hip/hip_bf16.h>
#include <math.h>

// ---------------- problem constants ----------------
#define BATCH   32
#define D_INP   2
#define LEN     4096      // L
#define HDIM    128       // H
#define NC      32        // N_STATE/2 complex modes
#define TCH     64        // chunk length T
#define NCHUNK  64        // L / TCH
#define NLAYERS 6
#define DOUT    2

typedef float v2f __attribute__((ext_vector_type(2)));
typedef float v4f __attribute__((ext_vector_type(4)));
typedef float v8f __attribute__((ext_vector_type(8)));

// fp32 WMMA: D(16x16,f32) = A(16x4,f32) * B(4x16,f32) + C
__device__ __forceinline__ v8f wmma4(v2f a, v2f b, v8f c) {
    return __builtin_amdgcn_wmma_f32_16x16x4_f32(
        /*neg_a=*/false, a, /*neg_b=*/false, b,
        /*c_mod=*/(short)0, c, /*reuse_a=*/false, /*reuse_b=*/false);
}

// ---- async global -> LDS copy (CDNA5 ASYNCcnt path), with safe fallback ----
#if __has_builtin(__builtin_amdgcn_global_load_async_to_lds_b128)
#define HAS_ASYNC_LDS 1
typedef int gv4i __attribute__((vector_size(16)));
typedef __attribute__((address_space(1))) gv4i as1_v4i;
typedef __attribute__((address_space(3))) gv4i as3_v4i;
#else
#define HAS_ASYNC_LDS 0
#endif

__device__ __forceinline__ void lds_copy16(const float* g, float* l) {
#if HAS_ASYNC_LDS
    __builtin_amdgcn_global_load_async_to_lds_b128(
        (as1_v4i*)g, (as3_v4i*)l, /*offset=*/0, /*cpol=*/0);
#else
    *(v4f*)l = *(const v4f*)g;
#endif
}

__device__ __forceinline__ void async_join() {
#if HAS_ASYNC_LDS
#if __has_builtin(__builtin_amdgcn_s_wait_asynccnt)
    __builtin_amdgcn_s_wait_asynccnt(0);
#else
    asm volatile("s_wait_asynccnt 0x0" ::: "memory");
#endif
#endif
    __syncthreads();
}

// ---------------- workspace layout (in floats) ----------------
#define OFF_H    ((size_t)0)                         // (B,H,L) activations
#define OFF_Y    (OFF_H + (size_t)BATCH*HDIM*LEN)    // post conv+gelu
#define OFF_Z    (OFF_Y + (size_t)BATCH*HDIM*LEN)    // post GLU
#define OFF_P    (OFF_Z + (size_t)BATCH*HDIM*LEN)    // [h][b][c][64] chunk summaries / carries
#define OFF_M    (OFF_P + (size_t)HDIM*BATCH*NCHUNK*64)   // [h][64][128] toeplitz+decay
#define OFF_V    (OFF_M + (size_t)HDIM*64*128)            // [h][64][64] vandermonde
#define OFF_AL   (OFF_V + (size_t)HDIM*64*64)             // lambda^T re (H*NC) then im (H*NC)
#define OFF_POOL (OFF_AL + (size_t)2*HDIM*NC)             // (B,H)

// ---------------- K0: encoder  h[b,h,l] = x[b,:,l] @ enc_W + enc_b ----------------
__global__ void k_encoder(const float* __restrict__ x, const float* __restrict__ encW,
                          const float* __restrict__ encb, float* __restrict__ hbuf) {
    size_t idx = (size_t)blockIdx.x * blockDim.x + threadIdx.x;   // ((b*H+h)*L + l)
    int l  = (int)(idx & (LEN - 1));
    size_t t = idx >> 12;
    int hh = (int)(t & (HDIM - 1));
    int bb = (int)(t >> 7);
    float v = x[(size_t)(bb * D_INP + 0) * LEN + l] * encW[hh]
            + x[(size_t)(bb * D_INP + 1) * LEN + l] * encW[HDIM + hh]
            + encb[hh];
    hbuf[idx] = v;
}

// ---------------- K1: per-(layer,h) discretization, kernels, matrices ----------------
__global__ void k_prep(const float* __restrict__ log_dt, const float* __restrict__ C_re,
                       const float* __restrict__ C_im,  const float* __restrict__ log_A_real,
                       const float* __restrict__ A_imag,
                       float* __restrict__ Mbuf, float* __restrict__ Vbuf,
                       float* __restrict__ Abuf, int layer) {
    int h = blockIdx.x;
    int t = threadIdx.x;                  // 0..63
    __shared__ float pwr[NC][TCH + 1];    // Re(lambda^j), j=0..64
    __shared__ float pwi[NC][TCH + 1];
    __shared__ float cer[NC], cei[NC];    // ZOH-discretized C
    __shared__ float Ksh[TCH];            // conv kernel K[j]

    if (t < NC) {
        int n = t;
        float dt  = expf(log_dt[(size_t)layer * HDIM + h]);
        size_t pidx = ((size_t)layer * HDIM + h) * NC + n;
        float Are = -expf(log_A_real[pidx]);
        float Aim = A_imag[pidx];
        float dr = Are * dt, di = Aim * dt;
        float er = expf(dr);
        float lr = er * cosf(di), li = er * sinf(di);     // lambda = exp(dtA)
        // (lambda-1)/A  (complex)
        float den = Are * Are + Aim * Aim;
        float fr = ((lr - 1.f) * Are + li * Aim) / den;
        float fi = (li * Are - (lr - 1.f) * Aim) / den;
        float cre = C_re[pidx], cim = C_im[pidx];
        cer[n] = cre * fr - cim * fi;
        cei[n] = cre * fi + cim * fr;
        // powers lambda^0 .. lambda^64
        float pr = 1.f, pi = 0.f;
        for (int j = 0; j <= TCH; j++) {
            pwr[n][j] = pr; pwi[n][j] = pi;
            float nr = pr * lr - pi * li;
            pi = pr * li + pi * lr;
            pr = nr;
        }
        // alpha = lambda^T
        Abuf[(size_t)h * NC + n]              = pwr[n][TCH];
        Abuf[(size_t)HDIM * NC + h * NC + n]  = pwi[n][TCH];
    }
    __syncthreads();
    // K[j] = 2 * Re( sum_n Ceff_n * lambda_n^j )
    {
        float s = 0.f;
        for (int n = 0; n < NC; n++)
            s += cer[n] * pwr[n][t] - cei[n] * pwi[n][t];
        Ksh[t] = 2.f * s;
    }
    __syncthreads();
    // M row t: [ lower-Toeplitz(K) (64) | 2Re(C*l^{t+1}) (32) | -2Im(C*l^{t+1}) (32) ]
    {
        float* Mr = Mbuf + ((size_t)h * TCH + t) * 128;
        for (int m = 0; m < TCH; m++)
            Mr[m] = (m <= t) ? Ksh[t - m] : 0.f;
        for (int n = 0; n < NC; n++) {
            float gr = cer[n] * pwr[n][t + 1] - cei[n] * pwi[n][t + 1];
            float gi = cer[n] * pwi[n][t + 1] + cei[n] * pwr[n][t + 1];
            Mr[64 + n] =  2.f * gr;
            Mr[96 + n] = -2.f * gi;
        }
    }
    // V rows: V[n][m] = Re(l^{T-1-m}), V[32+n][m] = Im(l^{T-1-m})
    if (t < NC) {
        int n = t;
        float* Vr0 = Vbuf + ((size_t)h * 64 + n) * TCH;
        float* Vr1 = Vbuf + ((size_t)h * 64 + 32 + n) * TCH;
        for (int m = 0; m < TCH; m++) {
            Vr0[m] = pwr[n][TCH - 1 - m];
            Vr1[m] = pwi[n][TCH - 1 - m];
        }
    }
}

// ---------------- K2: chunk summaries  P[h][b][c][:] = V_h @ u_chunk (WMMA) ----------------
// u-chunk tile staged in LDS: [c][k], column stride 68 floats (16B aligned, bank-spread)
__global__ void k_phase1(const float* __restrict__ hbuf, const float* __restrict__ Vbuf,
                         float* __restrict__ Pbuf) {
    __shared__ float su[16 * 68];
    int h = blockIdx.y;
    int j0 = blockIdx.x * 16;            // column tile over (b*NCHUNK + c) = 2048 cols
    int tid = threadIdx.x;
    int wave = tid >> 5;                 // row tile 0..3 (rows 0..63)
    int lane = tid & 31;
    int nidx = lane & 15;
    int khalf = (lane < 16) ? 0 : 2;
    int bb = j0 >> 6;
    int c0 = j0 & 63;
    const float* U = hbuf + ((size_t)bb * HDIM + h) * LEN;
    // stage 16 columns x 64 floats (256 b128 quads) via async global->LDS
    for (int q = tid; q < 256; q += 128) {
        int cc = q >> 4;
        int kq = (q & 15) * 4;
        lds_copy16(U + (size_t)(c0 + cc) * 64 + kq, &su[cc * 68 + kq]);
    }
    async_join();
    const float* A = Vbuf + ((size_t)h * 64 + wave * 16 + nidx) * TCH;
    v8f acc = {0.f,0.f,0.f,0.f,0.f,0.f,0.f,0.f};
    #pragma unroll
    for (int k = 0; k < TCH; k += 4) {
        v2f a; a.x = A[k + khalf];               a.y = A[k + khalf + 1];
        v2f b; b.x = su[nidx * 68 + k + khalf];  b.y = su[nidx * 68 + k + khalf + 1];
        acc = wmma4(a, b, acc);
    }
    int c = c0 + nidx;
    float* Pp = Pbuf + (((size_t)h * BATCH + bb) * NCHUNK + c) * 64;
    int rbase = wave * 16 + ((lane < 16) ? 0 : 8);
    #pragma unroll
    for (int r = 0; r < 8; r++) Pp[rbase + r] = acc[r];
}

// ---------------- K3: carry scan over chunks (in-place: P[c] <- state before chunk c) ---------
__global__ void k_scan(float* __restrict__ Pbuf, const float* __restrict__ Abuf) {
    int tid = blockIdx.x * blockDim.x + threadIdx.x;    // b*H*NC lanes
    int n  = tid & 31;
    int h  = (tid >> 5) & (HDIM - 1);
    int bb = tid >> 12;
    float ar = Abuf[(size_t)h * NC + n];
    float ai = Abuf[(size_t)HDIM * NC + h * NC + n];
    float sr = 0.f, si = 0.f;
    float* Pp = Pbuf + (((size_t)h * BATCH + bb) * NCHUNK) * 64;
    for (int c = 0; c < NCHUNK; c++) {
        float pr = Pp[c * 64 + n];
        float pi = Pp[c * 64 + 32 + n];
        Pp[c * 64 + n]      = sr;     // state entering chunk c
        Pp[c * 64 + 32 + n] = si;
        float nr = ar * sr - ai * si + pr;
        si = ar * si + ai * sr + pi;
        sr = nr;
    }
}

// ---------------- K4: y = gelu( M_h @ [u_chunk; carry] + D*u )  (WMMA) ----------------
__global__ void k_conv(const float* __restrict__ hbuf, const float* __restrict__ Mbuf,
                       const float* __restrict__ Pbuf, const float* __restrict__ Dp,
                       float* __restrict__ ybuf, int layer) {
    __shared__ float su[16 * 68];    // u chunk   [c][k0..63]
    __shared__ float sp[16 * 68];    // carries   [c][r0..63]
    int h = blockIdx.y;
    int j0 = blockIdx.x * 16;
    int tid = threadIdx.x;
    int wave = tid >> 5;
    int lane = tid & 31;
    int nidx = lane & 15;
    int khalf = (lane < 16) ? 0 : 2;
    int bb = j0 >> 6;
    int c0 = j0 & 63;
    const float* U  = hbuf + ((size_t)bb * HDIM + h) * LEN;
    const float* Pb = Pbuf + (((size_t)h * BATCH + bb) * NCHUNK) * 64;
    // stage u tile + carry tile (512 quads total)
    for (int q = tid; q < 512; q += 128) {
        int cc = (q >> 4) & 15;
        int kq = (q & 15) * 4;
        if (q < 256) lds_copy16(U  + (size_t)(c0 + cc) * 64 + kq, &su[cc * 68 + kq]);
        else         lds_copy16(Pb + (size_t)(c0 + cc) * 64 + kq, &sp[cc * 68 + kq]);
    }
    async_join();
    const float* A = Mbuf + ((size_t)h * TCH + wave * 16 + nidx) * 128;
    v8f acc = {0.f,0.f,0.f,0.f,0.f,0.f,0.f,0.f};
    #pragma unroll
    for (int k = 0; k < 64; k += 4) {        // Toeplitz part: rows = u within chunk
        v2f a; a.x = A[k + khalf];               a.y = A[k + khalf + 1];
        v2f b; b.x = su[nidx * 68 + k + khalf];  b.y = su[nidx * 68 + k + khalf + 1];
        acc = wmma4(a, b, acc);
    }
    #pragma unroll
    for (int k = 64; k < 128; k += 4) {      // decay part: rows = carry state (re|im)
        v2f a; a.x = A[k + khalf];                    a.y = A[k + khalf + 1];
        v2f b; b.x = sp[nidx * 68 + k - 64 + khalf];  b.y = sp[nidx * 68 + k - 64 + khalf + 1];
        acc = wmma4(a, b, acc);
    }
    float Dh = Dp[(size_t)layer * HDIM + h];
    float* Yp = ybuf + ((size_t)bb * HDIM + h) * LEN;
    int c = c0 + nidx;
    int tbase = wave * 16 + ((lane < 16) ? 0 : 8);
    #pragma unroll
    for (int r = 0; r < 8; r++) {
        int t = tbase + r;
        int l = c * 64 + t;
        float uv = su[nidx * 68 + t];
        float v = acc[r] + Dh * uv;
        Yp[l] = 0.5f * v * (1.f + erff(v * 0.70710678f));   // exact GeLU
    }
}

// ---------------- K5: pointwise conv H->2H + GLU fused  (WMMA) ----------------
__global__ void k_pointwise(const float* __restrict__ ybuf, const float* __restrict__ outW,
                            const float* __restrict__ outb, float* __restrict__ zbuf, int layer) {
    __shared__ float sy[128 * 16];       // [k][16 cols], row = 64B (conflict-free)
    int j0 = blockIdx.x * 16;            // column over (b*L) = 131072
    int tid = threadIdx.x;
    int wave = tid >> 5;                 // o-row tile 0..7
    int lane = tid & 31;
    int nidx = lane & 15;
    int khalf = (lane < 16) ? 0 : 2;
    int bb = j0 >> 12;
    int l0 = j0 & (LEN - 1);
    const float* Y = ybuf + (size_t)bb * HDIM * LEN;
    // stage B tile: 128 rows x 16 cols = 512 quads
    for (int q = tid; q < 512; q += 256) {
        int k = q >> 2;
        int off = (q & 3) * 4;
        lds_copy16(Y + (size_t)k * LEN + l0 + off, &sy[k * 16 + off]);
    }
    async_join();
    const float* Wl = outW + (size_t)layer * 2 * HDIM * HDIM;
    int rowA = wave * 16 + nidx;
    v8f accA = {0.f,0.f,0.f,0.f,0.f,0.f,0.f,0.f};
    v8f accG = {0.f,0.f,0.f,0.f,0.f,0.f,0.f,0.f};
    #pragma unroll 8
    for (int k = 0; k < HDIM; k += 4) {
        v2f b;  b.x  = sy[(k + khalf) * 16 + nidx];
                b.y  = sy[(k + khalf + 1) * 16 + nidx];
        v2f aA; aA.x = Wl[(size_t)rowA * HDIM + k + khalf];
                aA.y = Wl[(size_t)rowA * HDIM + k + khalf + 1];
        v2f aG; aG.x = Wl[(size_t)(rowA + HDIM) * HDIM + k + khalf];
                aG.y = Wl[(size_t)(rowA + HDIM) * HDIM + k + khalf + 1];
        accA = wmma4(aA, b, accA);
        accG = wmma4(aG, b, accG);
    }
    const float* bvec = outb + (size_t)layer * 2 * HDIM;
    int l = l0 + nidx;
    int obase = wave * 16 + ((lane < 16) ? 0 : 8);
    #pragma unroll
    for (int r = 0; r < 8; r++) {
        int o = obase + r;
        float a = accA[r] + bvec[o];
        float g = accG[r] + bvec[HDIM + o];
        float z = a * (1.f / (1.f + expf(-g)));
        zbuf[((size_t)bb * HDIM + o) * LEN + l] = z;
    }
}

// ---------------- K6: residual + LayerNorm over channel dim ----------------
__global__ void k_residln(const float* __restrict__ zbuf, float* __restrict__ hbuf,
                          const float* __restrict__ lng, const float* __restrict__ lnb, int layer) {
    int idx = blockIdx.x * blockDim.x + threadIdx.x;   // b*L + l
    int bb = idx >> 12;
    int l  = idx & (LEN - 1);
    size_t base = (size_t)bb * HDIM * LEN + l;
    float s = 0.f, sq = 0.f;
    for (int hh = 0; hh < HDIM; hh++) {
        float v = zbuf[base + (size_t)hh * LEN] + hbuf[base + (size_t)hh * LEN];
        s += v; sq += v * v;
    }
    float m = s * (1.f / HDIM);
    float var = fmaxf(sq * (1.f / HDIM) - m * m, 0.f);
    float inv = rsqrtf(var + 1e-5f);
    const float* g = lng + (size_t)layer * HDIM;
    const float* b = lnb + (size_t)layer * HDIM;
    for (int hh = 0; hh < HDIM; hh++) {
        float v = zbuf[base + (size_t)hh * LEN] + hbuf[base + (size_t)hh * LEN];
        hbuf[base + (size_t)hh * LEN] = (v - m) * inv * g[hh] + b[hh];
    }
}

// ---------------- K7: mean pool over L (one wave32 per (b,h)) ----------------
__global__ void k_pool(const float* __restrict__ hbuf, float* __restrict__ pooled) {
    int w = (blockIdx.x * blockDim.x + threadIdx.x) >> 5;
    int lane = threadIdx.x & 31;
    if (w >= BATCH * HDIM) return;
    const float* p = hbuf + (size_t)w * LEN;
    float s = 0.f;
    for (int j = lane; j < LEN; j += 32) s += p[j];
    #pragma unroll
    for (int off = 16; off > 0; off >>= 1) s += __shfl_xor(s, off, 32);
    if (lane == 0) pooled[w] = s * (1.f / LEN);
}

// ---------------- K8: head GEMV ----------------
__global__ void k_head(const float* __restrict__ pooled, const float* __restrict__ headW,
                       const float* __restrict__ headb, float* __restrict__ out) {
    int t = threadIdx.x;
    if (t >= BATCH * DOUT) return;
    int bb = t >> 1, d = t & 1;
    float acc = headb[d];
    for (int hh = 0; hh < HDIM; hh++)
        acc += pooled[(size_t)bb * HDIM + hh] * headW[(size_t)hh * DOUT + d];
    out[t] = acc;
}

// ---------------- launch ----------------
extern "C" void kernel_launch(void* const* d_in, const int* in_sizes, int n_in,
                              void* d_out, int out_size, void* d_ws, size_t ws_size,
                              hipStream_t stream) {
    const float* x     = (const float*)d_in[0];
    const float* encW  = (const float*)d_in[1];
    const float* encb  = (const float*)d_in[2];
    const float* logdt = (const float*)d_in[3];
    const float* Cre   = (const float*)d_in[4];
    const float* Cim   = (const float*)d_in[5];
    const float* logAr = (const float*)d_in[6];
    const float* Aim   = (const float*)d_in[7];
    const float* Dp    = (const float*)d_in[8];
    const float* outW  = (const float*)d_in[9];
    const float* outb  = (const float*)d_in[10];
    const float* lng   = (const float*)d_in[11];
    const float* lnb   = (const float*)d_in[12];
    const float* headW = (const float*)d_in[13];
    const float* headb = (const float*)d_in[14];

    float* ws    = (float*)d_ws;
    float* hbuf  = ws + OFF_H;
    float* ybuf  = ws + OFF_Y;
    float* zbuf  = ws + OFF_Z;
    float* Pbuf  = ws + OFF_P;
    float* Mbuf  = ws + OFF_M;
    float* Vbuf  = ws + OFF_V;
    float* Abuf  = ws + OFF_AL;
    float* pool  = ws + OFF_POOL;

    // encoder: B*H*L threads
    k_encoder<<<(BATCH * HDIM * LEN) / 256, 256, 0, stream>>>(x, encW, encb, hbuf);

    for (int i = 0; i < NLAYERS; i++) {
        k_prep<<<HDIM, 64, 0, stream>>>(logdt, Cre, Cim, logAr, Aim, Mbuf, Vbuf, Abuf, i);
        k_phase1<<<dim3(128, HDIM), 128, 0, stream>>>(hbuf, Vbuf, Pbuf);
        k_scan<<<(BATCH * HDIM * NC) / 256, 256, 0, stream>>>(Pbuf, Abuf);
        k_conv<<<dim3(128, HDIM), 128, 0, stream>>>(hbuf, Mbuf, Pbuf, Dp, ybuf, i);
        k_pointwise<<<(BATCH * LEN) / 16, 256, 0, stream>>>(ybuf, outW, outb, zbuf, i);
        k_residln<<<(BATCH * LEN) / 256, 256, 0, stream>>>(zbuf, hbuf, lng, lnb, i);
    }

    k_pool<<<(BATCH * HDIM * 32) / 256, 256, 0, stream>>>(hbuf, pool);
    k_head<<<1, 64, 0, stream>>>(pool, headW, headb, (float*)d_out);
}